// GNNModel_13451837571701
// MI455X (gfx1250) — compile-verified
//
#include <hip/hip_runtime.h>
#include <hip/hip_bf16.h>

typedef float v2f __attribute__((ext_vector_type(2)));
typedef float v8f __attribute__((ext_vector_type(8)));
typedef int   v4i_ls __attribute__((vector_size(16)));   // matches builtin's param type

#define N_NODES 20000
#define N_EDGES 400000
#define NODE_D  256
#define HIDDEN  256
#define EDGE_D  16
#define OUT_D   128
#define NTILES  (N_EDGES / 16)   // 25000
#define NBLK_B  256

#if defined(__has_builtin)
#  if __has_builtin(__builtin_amdgcn_global_load_async_to_lds_b128)
#    define HAVE_ASYNC_LDS 1
#  endif
#endif
#ifndef HAVE_ASYNC_LDS
#  define HAVE_ASYNC_LDS 0
#endif

// ---------------------------------------------------------------------------
// Kernel A: Out[20000x256] = X[20000x256] @ W[256x256] (+ bias)
// One block = 32 rows x 256 cols; 8 waves; each wave: 2 row-tiles x 2 col-tiles
// (B-frags reused across the two row-tiles -> 4 global b32 loads per 4 WMMAs).
// x tile staged to LDS via async-to-LDS when available.
// ---------------------------------------------------------------------------
__global__ __launch_bounds__(256) void gemm_node(
    const float* __restrict__ X, const float* __restrict__ W,
    const float* __restrict__ bias, float* __restrict__ Out) {
  __shared__ float xs[32 * 260];                 // pad 256->260: conflict-free b64 reads
  const int tid = threadIdx.x;
  const int row0 = blockIdx.x * 32;

  // stage the 32x256 A tile (2048 float4s, 8 per thread)
#pragma unroll
  for (int i = 0; i < 8; ++i) {
    int f = tid + 256 * i;                       // float4 index 0..2047
    int r = f >> 6;
    int c4 = (f & 63) << 2;
    const size_t gidx = (size_t)(row0 + r) * NODE_D + c4;
    const int lidx = r * 260 + c4;
#if HAVE_ASYNC_LDS
    __builtin_amdgcn_global_load_async_to_lds_b128(
        (__attribute__((address_space(1))) v4i_ls*)(X + gidx),
        (__attribute__((address_space(3))) v4i_ls*)&xs[lidx], 0, 0);
#else
    *(float4*)&xs[lidx] = *(const float4*)(X + gidx);
#endif
  }
#if HAVE_ASYNC_LDS
  asm volatile("s_wait_asynccnt 0" ::: "memory");
#endif
  __syncthreads();

  const int lane = tid & 31;
  const int wave = tid >> 5;
  const int hi   = lane >> 4;                    // 0: K sub {0,1}, 1: K sub {2,3}
  const int ln   = lane & 15;
  const int col0 = (wave)     * 16 + ln;
  const int col1 = (wave + 8) * 16 + ln;

  v8f c00 = {0.f, 0.f, 0.f, 0.f, 0.f, 0.f, 0.f, 0.f};
  v8f c01 = c00, c10 = c00, c11 = c00;

#pragma unroll 8
  for (int k = 0; k < NODE_D; k += 4) {
    v2f a0 = *(const v2f*)&xs[ln        * 260 + k + 2 * hi];   // rows 0-15
    v2f a1 = *(const v2f*)&xs[(16 + ln) * 260 + k + 2 * hi];   // rows 16-31
    v2f b0, b1;
    b0.x = W[(size_t)(k + 2 * hi)     * HIDDEN + col0];
    b0.y = W[(size_t)(k + 2 * hi + 1) * HIDDEN + col0];
    b1.x = W[(size_t)(k + 2 * hi)     * HIDDEN + col1];
    b1.y = W[(size_t)(k + 2 * hi + 1) * HIDDEN + col1];
    c00 = __builtin_amdgcn_wmma_f32_16x16x4_f32(false, a0, false, b0, (short)0, c00, false, false);
    c01 = __builtin_amdgcn_wmma_f32_16x16x4_f32(false, a0, false, b1, (short)0, c01, false, false);
    c10 = __builtin_amdgcn_wmma_f32_16x16x4_f32(false, a1, false, b0, (short)0, c10, false, false);
    c11 = __builtin_amdgcn_wmma_f32_16x16x4_f32(false, a1, false, b1, (short)0, c11, false, false);
  }

  const float add0 = bias ? bias[col0] : 0.f;
  const float add1 = bias ? bias[col1] : 0.f;
#pragma unroll
  for (int j = 0; j < 8; ++j) {
    const int ra = row0 + j + 8 * hi;            // row-tile 0
    const int rb = ra + 16;                      // row-tile 1
    Out[(size_t)ra * HIDDEN + col0] = c00[j] + add0;
    Out[(size_t)ra * HIDDEN + col1] = c01[j] + add1;
    Out[(size_t)rb * HIDDEN + col0] = c10[j] + add0;
    Out[(size_t)rb * HIDDEN + col1] = c11[j] + add1;
  }
}

// ---------------------------------------------------------------------------
// Kernel B: per 16-edge tile, per batch:
//   h = relu( (p[src]+q[dst])  +  attrs(16x16) @ W_attr(16x256) )   [WMMA]
// summed over edges into per-block partials (deterministic tree reduction).
// ---------------------------------------------------------------------------
__global__ __launch_bounds__(256) void edge_accum(
    const float* __restrict__ P, const float* __restrict__ Q,
    const int* __restrict__ EI, const float* __restrict__ EA,
    const float* __restrict__ Wattr, float* __restrict__ Partial) {
  __shared__ float wT[256 * 18];                 // W_attr transposed [col][k], pad 16->18
  __shared__ float slab[8 * 4 * 256];            // per-wave partial sums [wave][b][col]

  const int tid = threadIdx.x;
  {
    const int c = tid;                           // one column per thread
#pragma unroll
    for (int k = 0; k < EDGE_D; ++k) wT[c * 18 + k] = Wattr[k * HIDDEN + c];
  }
  __syncthreads();

  const int lane = tid & 31;
  const int wave = tid >> 5;
  const int hi   = lane >> 4;
  const int ln   = lane & 15;
  const int gw   = blockIdx.x * 8 + wave;
  const int nw   = NBLK_B * 8;

  float accs[4][16];
#pragma unroll
  for (int b = 0; b < 4; ++b)
#pragma unroll
    for (int ct = 0; ct < 16; ++ct) accs[b][ct] = 0.f;

  for (int t = gw; t < NTILES; t += nw) {
    const int e0 = t * 16;

    // prefetch next tile's attr rows + indices into cache while we compute
    const int tn = t + nw;
    if (tn < NTILES) {
      __builtin_prefetch(EI + tn * 16, 0, 1);
      __builtin_prefetch(EI + N_EDGES + tn * 16, 0, 1);
#pragma unroll
      for (int b = 0; b < 4; ++b)
        __builtin_prefetch(EA + ((size_t)b * N_EDGES + tn * 16 + ln) * EDGE_D, 0, 1);
    }

    int srcs[16], dsts[16];
#pragma unroll
    for (int r = 0; r < 16; ++r) {
      srcs[r] = EI[e0 + r];
      dsts[r] = EI[N_EDGES + e0 + r];
    }
    // per-lane C-layout row bases: VGPR j holds row j (lanes<16) / row j+8 (lanes>=16)
    int rbP[8], rbQ[8];
#pragma unroll
    for (int j = 0; j < 8; ++j) {
      rbP[j] = (hi ? srcs[j + 8] : srcs[j]) * HIDDEN;
      rbQ[j] = (hi ? dsts[j + 8] : dsts[j]) * HIDDEN;
    }
    // A fragments: attrs[b][edge=ln (+ M from lane)][k], f32 16x4 chain layout
    v2f af[4][4];
    const int eL = e0 + ln;
#pragma unroll
    for (int b = 0; b < 4; ++b) {
      const float* ap = EA + ((size_t)b * N_EDGES + eL) * EDGE_D + 2 * hi;
#pragma unroll
      for (int kc = 0; kc < 4; ++kc) af[b][kc] = *(const v2f*)(ap + 4 * kc);
    }

#pragma unroll 2
    for (int ct = 0; ct < 16; ++ct) {
      const int col = ct * 16 + ln;
      v2f wf[4];
#pragma unroll
      for (int kc = 0; kc < 4; ++kc)
        wf[kc] = *(const v2f*)&wT[col * 18 + 4 * kc + 2 * hi];

      float bse[8];
#pragma unroll
      for (int j = 0; j < 8; ++j) bse[j] = P[rbP[j] + col] + Q[rbQ[j] + col];

#pragma unroll
      for (int b = 0; b < 4; ++b) {
        v8f c = {bse[0], bse[1], bse[2], bse[3], bse[4], bse[5], bse[6], bse[7]};
#pragma unroll
        for (int kc = 0; kc < 4; ++kc)
          c = __builtin_amdgcn_wmma_f32_16x16x4_f32(false, af[b][kc], false, wf[kc],
                                                    (short)0, c, false, false);
        float s = 0.f;
#pragma unroll
        for (int j = 0; j < 8; ++j) s += fmaxf(c[j], 0.f);
        accs[b][ct] += s;
      }
    }
  }

  // lane l holds rows 0-7, lane l+16 rows 8-15 of same cols: pair-add, stash.
#pragma unroll
  for (int b = 0; b < 4; ++b)
#pragma unroll
    for (int ct = 0; ct < 16; ++ct) {
      float v = accs[b][ct];
      v += __shfl_xor(v, 16, 32);
      if (hi == 0) slab[wave * 1024 + b * 256 + ct * 16 + ln] = v;
    }
  __syncthreads();

  for (int i = tid; i < 1024; i += 256) {
    float s = 0.f;
#pragma unroll
    for (int w = 0; w < 8; ++w) s += slab[w * 1024 + i];
    Partial[(size_t)blockIdx.x * 1024 + i] = s;
  }
}

// ---------------------------------------------------------------------------
// Kernel C: reduce partials, mean, then (4x256)@(256x128)+b -> d_out (4x128)
// ---------------------------------------------------------------------------
__global__ __launch_bounds__(1024) void finalize(
    const float* __restrict__ Partial, const float* __restrict__ Wg,
    const float* __restrict__ bg, float* __restrict__ Out) {
  __shared__ float gf[1024];
  const int tid = threadIdx.x;
  {
    float s = 0.f;
    for (int blk = 0; blk < NBLK_B; ++blk) s += Partial[(size_t)blk * 1024 + tid];
    gf[tid] = s * (1.0f / (float)N_EDGES);
  }
  __syncthreads();
  if (tid < 4 * OUT_D) {
    const int b = tid >> 7, o = tid & 127;
    float s = bg[o];
    const float* g = &gf[b * 256];
#pragma unroll 8
    for (int c = 0; c < HIDDEN; ++c) s = fmaf(g[c], Wg[c * OUT_D + o], s);
    Out[tid] = s;
  }
}

extern "C" void kernel_launch(void* const* d_in, const int* in_sizes, int n_in,
                              void* d_out, int out_size, void* d_ws, size_t ws_size,
                              hipStream_t stream) {
  const float* x  = (const float*)d_in[0];
  const int*   ei = (const int*)d_in[1];
  const float* ea = (const float*)d_in[2];
  const float* We = (const float*)d_in[3];
  const float* be = (const float*)d_in[4];
  const float* Wg = (const float*)d_in[5];
  const float* bg = (const float*)d_in[6];
  float* out = (float*)d_out;

  float* ws = (float*)d_ws;
  float* P    = ws;                                    // 20000*256
  float* Q    = ws + (size_t)N_NODES * HIDDEN;         // 20000*256
  float* Part = ws + (size_t)2 * N_NODES * HIDDEN;     // NBLK_B*1024

  gemm_node<<<N_NODES / 32, 256, 0, stream>>>(x, We, be, P);                       // p = x@W_src + b
  gemm_node<<<N_NODES / 32, 256, 0, stream>>>(x, We + 256 * 256, nullptr, Q);      // q = x@W_dst
  edge_accum<<<NBLK_B, 256, 0, stream>>>(P, Q, ei, ea, We + 512 * 256, Part);
  finalize<<<1, 1024, 0, stream>>>(Part, Wg, bg, out);
}